// Att_21973052686618
// MI455X (gfx1250) — compile-verified
//
#include <hip/hip_runtime.h>

typedef float v2f __attribute__((ext_vector_type(2)));
typedef float v8f __attribute__((ext_vector_type(8)));
typedef unsigned int u32x4 __attribute__((ext_vector_type(4)));
typedef int i32x4 __attribute__((ext_vector_type(4)));
typedef int i32x8 __attribute__((ext_vector_type(8)));

#define N_SEQ 2048
#define D_IN  512

#ifndef __has_builtin
#define __has_builtin(x) 0
#endif

#if __has_builtin(__builtin_amdgcn_global_load_async_to_lds_b128)
#define HAVE_ASYNC_LDS 1
#else
#define HAVE_ASYNC_LDS 0
#endif

#if __has_builtin(__builtin_amdgcn_tensor_load_to_lds) && \
    __has_builtin(__builtin_amdgcn_s_wait_tensorcnt)
#define HAVE_TDM 1
#else
#define HAVE_TDM 0
#endif

#define AS1 __attribute__((address_space(1)))
#define AS3 __attribute__((address_space(3)))

// ---- async global->LDS (per-lane 16B), ASYNCcnt tracked ---------------------
// Signature (from diagnostics): (AS1 int4* global_src, AS3 int4* lds_dst, Ii, Ii)
__device__ __forceinline__ void cp_async16(void* lds, const void* g) {
#if HAVE_ASYNC_LDS
  __builtin_amdgcn_global_load_async_to_lds_b128(
      (AS1 i32x4*)(unsigned long long)g,
      (AS3 i32x4*)(unsigned)(unsigned long long)lds, 0, 0);
#else
  *(float4*)lds = *(const float4*)g;
#endif
}
__device__ __forceinline__ void cp_wait() {
#if HAVE_ASYNC_LDS
#if __has_builtin(__builtin_amdgcn_s_wait_asynccnt)
  __builtin_amdgcn_s_wait_asynccnt(0);
#else
  asm volatile("s_wait_asynccnt 0" ::: "memory");
#endif
#endif
}

// ---- TDM 2D tile load: D# per CDNA5 ISA 8.3-8.5 (4B elements) ---------------
// 6-arg toolchain form: (u32x4 g0, i32x8 g1, i32x4 g2, i32x4 g3, i32x8, i32 cpol)
#if HAVE_TDM
__device__ __forceinline__ void tdm_load_2d(void* lds, const void* g,
                                            unsigned tensor_d0, unsigned tensor_d1,
                                            unsigned tile_d0, unsigned tile_d1,
                                            unsigned stride0) {
  unsigned long long ga = (unsigned long long)g;
  u32x4 g0;
  g0[0] = 1u;                                         // count=1, user descriptor
  g0[1] = (unsigned)(unsigned long long)lds;          // lds_addr (bytes)
  g0[2] = (unsigned)ga;                               // global_addr[31:0]
  g0[3] = (unsigned)((ga >> 32) & 0x1FFFFFFull) | (2u << 30);  // addr[56:32]|type=2
  i32x8 g1;
  g1[0] = 0x00020000;                                 // data_size = 4B
  g1[1] = (int)((tensor_d0 & 0xFFFFu) << 16);         // tensor_dim0[15:0]
  g1[2] = (int)(((tensor_d0 >> 16) & 0xFFFFu) | ((tensor_d1 & 0xFFFFu) << 16));
  g1[3] = (int)(((tensor_d1 >> 16) & 0xFFFFu) | ((tile_d0 & 0xFFFFu) << 16));
  g1[4] = (int)(tile_d1 & 0xFFFFu);                   // tile_dim1 (0 = 1D)
  g1[5] = (int)stride0;                               // tensor_dim0_stride[31:0]
  g1[6] = 0;
  g1[7] = 0;
  i32x4 z4 = {0, 0, 0, 0};
  i32x8 z8 = {0, 0, 0, 0, 0, 0, 0, 0};
  __builtin_amdgcn_tensor_load_to_lds(g0, g1, z4, z4, z8, 0);
}
#define TDM_WAIT() __builtin_amdgcn_s_wait_tensorcnt(0)
#endif

// D = A(16x4 f32) * B(4x16 f32) + C(16x16 f32)
__device__ __forceinline__ v8f wmma4(v2f a, v2f b, v8f c) {
  return __builtin_amdgcn_wmma_f32_16x16x4_f32(false, a, false, b, (short)0, c, false, false);
}

// ---------------- kernel 0: init workspace (zeros + V ones-row) --------------
// layout (floats): qbuf[0,512K) kbuf[512K,1M) vbuf_dm[1M,2M) cbuf[2M,2.5M)
// vbuf_dm is dim-major: [bh 32][dim 16][n 2048]; dim row 5 = 1.0 (softmax denom)
#define WS_FLOATS 2621440
#define VOFF      1048576
__global__ void init_ws_kernel(float* __restrict__ ws) {
  int i = blockIdx.x * blockDim.x + threadIdx.x;
  if (i >= WS_FLOATS) return;
  float v = 0.f;
  if (i >= VOFF && i < VOFF + 1048576) {
    int r = (i - VOFF) >> 11;                 // bh*16 + dim
    if ((r & 15) == 5) v = 1.f;
  }
  ws[i] = v;
}

// ---------------- kernel 1: fused QKV projection -----------------------------
__global__ __launch_bounds__(256) void qkv_kernel(
    const float* __restrict__ x,
    const float* __restrict__ Wq, const float* __restrict__ bq,
    const float* __restrict__ Wk, const float* __restrict__ bk,
    const float* __restrict__ Wv, const float* __restrict__ bv,
    float* __restrict__ qbuf, float* __restrict__ kbuf, float* __restrict__ vbuf) {
  __shared__ float xT[128][64];    // 32 KB, async-staged
  __shared__ float wTt[32][64];    // 8 KB, transposed [col][k] -> b64 fragments
  const int t = threadIdx.x;
  const int lane = t & 31, w = t >> 5;
  const int hi = lane >> 4, ln = lane & 15;
  const int R0 = blockIdx.x * 128;

  v8f acc0 = {}, acc1 = {};
  for (int kk = 0; kk < 8; ++kk) {
    #pragma unroll
    for (int i = 0; i < 8; ++i) {            // 2048 x-float4s via async DMA
      int idx = t + 256 * i;
      int row = idx >> 4, c4 = idx & 15;
      cp_async16(&xT[row][c4 * 4],
                 x + (size_t)(R0 + row) * D_IN + kk * 64 + c4 * 4);
    }
    #pragma unroll
    for (int i = 0; i < 8; ++i) {            // [Wq|Wk|Wv] slice, transposed
      int idx = t + 256 * i;
      int kr = idx >> 5, c = idx & 31;
      int kg = kk * 64 + kr;
      float wv = 0.f;
      if (c < 10)      wv = Wq[kg * 10 + c];
      else if (c < 20) wv = Wk[kg * 10 + (c - 10)];
      else if (c < 30) wv = Wv[kg * 10 + (c - 20)];
      wTt[c][kr] = wv;
    }
    cp_wait();
    __syncthreads();
    const int m0 = w * 16;
    #pragma unroll
    for (int k4 = 0; k4 < 16; ++k4) {
      int kb = k4 * 4 + hi * 2;
      v2f a;  a.x  = xT[m0 + ln][kb];       a.y  = xT[m0 + ln][kb + 1];
      v2f b0; b0.x = wTt[ln][kb];           b0.y = wTt[ln][kb + 1];
      v2f b1; b1.x = wTt[ln + 16][kb];      b1.y = wTt[ln + 16][kb + 1];
      acc0 = wmma4(a, b0, acc0);
      acc1 = wmma4(a, b1, acc1);
    }
    __syncthreads();
  }
  // epilogue: bias + scatter (q/k row-major stride-8; v dim-major)
  int c0 = ln, c1 = ln + 16;
  float bias0 = (c0 < 10) ? bq[c0] : bk[c0 - 10];
  float bias1 = (c1 < 20) ? bk[c1 - 10] : ((c1 < 30) ? bv[c1 - 20] : 0.f);
  #pragma unroll
  for (int r = 0; r < 8; ++r) {
    int rowg = R0 + w * 16 + hi * 8 + r;
    int b = rowg >> 11, n = rowg & 2047;
    float v0 = acc0[r] + bias0;
    if (c0 < 10) {
      int head = c0 / 5, d = c0 % 5;
      qbuf[(((size_t)(b * 2 + head)) * N_SEQ + n) * 8 + d] = v0;
    } else {
      int kc = c0 - 10, head = kc / 5, d = kc % 5;
      kbuf[(((size_t)(b * 2 + head)) * N_SEQ + n) * 8 + d] = v0;
    }
    float v1 = acc1[r] + bias1;
    if (c1 < 20) {
      int kc = c1 - 10, head = kc / 5, d = kc % 5;
      kbuf[(((size_t)(b * 2 + head)) * N_SEQ + n) * 8 + d] = v1;
    } else if (c1 < 30) {
      int vc = c1 - 20, head = vc / 5, d = vc % 5;
      vbuf[(((size_t)(b * 2 + head) * 16 + d)) * N_SEQ + n] = v1;
    }
  }
}

// ---------------- kernel 2: streaming causal attention -----------------------
__global__ __launch_bounds__(256) void attn_kernel(
    const float* __restrict__ qbuf, const float* __restrict__ kbuf,
    const float* __restrict__ vbuf, float* __restrict__ cbuf) {
  __shared__ float Kt[64][8];        // 2 KB, [key][dim] (contiguous copy)
  __shared__ float Vtt[16][64];      // 4 KB, [dim][key] (TDM 2D tile)
  __shared__ float Pt[8][16][16];    // 8 KB, per-wave C->A transpose tile
  const int t = threadIdx.x, lane = t & 31, w = t >> 5;
  const int hi = lane >> 4, ln = lane & 15;
  const int qb = blockIdx.x & 15;
  const int bh = blockIdx.x >> 4;
  const int b = bh >> 1, h = bh & 1;
  const float* qB = qbuf + (size_t)bh * N_SEQ * 8;
  const float* kB = kbuf + (size_t)bh * N_SEQ * 8;
  const float* vB = vbuf + (size_t)bh * 16 * N_SEQ;   // dim-major
  const int m0 = qb * 128 + w * 16;
  const float scale = 0.44721359549995794f;           // 1/sqrt(5)

  int lr = m0 + ln;
  v2f aq0, aq1;
  aq0.x = qB[lr * 8 + hi * 2] * scale;
  aq0.y = qB[lr * 8 + hi * 2 + 1] * scale;
  aq1.x = qB[lr * 8 + 4 + hi * 2] * scale;
  aq1.y = qB[lr * 8 + 4 + hi * 2 + 1] * scale;

  v8f o = {};
  const int nStage = (qb + 1) * 2;
  for (int s = 0; s < nStage; ++s) {
    int kbase = s * 64;
#if HAVE_TDM
    if (w == 0) {   // one TDM engine request per tile, TENSORcnt tracked
      tdm_load_2d(&Kt[0][0],  kB + (size_t)kbase * 8, 512, 1, 512, 0, 512);
      tdm_load_2d(&Vtt[0][0], vB + kbase, N_SEQ, 16, 64, 16, N_SEQ);
      TDM_WAIT();
    }
#else
    if (t < 128)
      cp_async16(((float*)Kt) + t * 4, kB + (size_t)kbase * 8 + t * 4);
    {
      int d = t >> 4, c4 = (t & 15) * 4;
      cp_async16(&Vtt[d][c4], vB + (size_t)d * N_SEQ + kbase + c4);
    }
    cp_wait();
#endif
    __syncthreads();
    for (int jb = 0; jb < 4; ++jb) {
      int key = jb * 16 + ln;
      v2f bk0, bk1;
      bk0.x = Kt[key][hi * 2];     bk0.y = Kt[key][hi * 2 + 1];
      bk1.x = Kt[key][4 + hi * 2]; bk1.y = Kt[key][4 + hi * 2 + 1];
      v8f sA = {};
      sA = wmma4(aq0, bk0, sA);
      sA = wmma4(aq1, bk1, sA);
      int ki = kbase + jb * 16 + ln;
      #pragma unroll
      for (int r = 0; r < 8; ++r) {
        int qi = m0 + hi * 8 + r;
        float sv = __sinf(sA[r]);
        float p = (ki <= qi) ? __expf(sv) : 0.f;   // causal mask + exp
        Pt[w][hi * 8 + r][ln] = p;
      }
      asm volatile("s_wait_dscnt 0" ::: "memory");  // cross-lane LDS RAW
      #pragma unroll
      for (int c = 0; c < 4; ++c) {                 // O += P(16x16) * V(16x16)
        int kr = c * 4 + hi * 2;
        v2f ap;  ap.x  = Pt[w][ln][kr];           ap.y  = Pt[w][ln][kr + 1];
        v2f bv2; bv2.x = Vtt[ln][jb * 16 + kr];   bv2.y = Vtt[ln][jb * 16 + kr + 1];
        o = wmma4(ap, bv2, o);
      }
    }
    __syncthreads();
  }
  int srcLane = (hi << 4) | 5;                      // O[:,5] = sum(exp)
  #pragma unroll
  for (int r = 0; r < 8; ++r) {
    float denom = __shfl(o[r], srcLane, 32);
    if (ln < 5) {
      int qi = m0 + hi * 8 + r;
      cbuf[((size_t)b * N_SEQ + qi) * 16 + h * 5 + ln] = o[r] / denom;
    }
  }
}

// ---------------- kernel 3: MLP head (10 -> 20 leaky_relu -> 512) ------------
__global__ __launch_bounds__(256) void mlp_kernel(
    const float* __restrict__ cbuf,
    const float* __restrict__ Wm1, const float* __restrict__ bm1,
    const float* __restrict__ Wm2, const float* __restrict__ bm2,
    float* __restrict__ out) {
  __shared__ float W1t[32][16];    // 2 KB transposed [col][k]
  __shared__ float W2[20][512];    // 40 KB row-major (TDM-staged, contiguous)
  __shared__ float b2s[512];
  __shared__ float b1s[32];
  __shared__ float T[8][16][32];   // 16 KB per-wave h1 transpose tile
  const int t = threadIdx.x, lane = t & 31, w = t >> 5;
  const int hi = lane >> 4, ln = lane & 15;
  const int R0 = blockIdx.x * 128;

  #pragma unroll
  for (int i = 0; i < 2; ++i) {
    int idx = t + 256 * i;
    int c = idx >> 4, r = idx & 15;
    W1t[c][r] = (r < 10 && c < 20) ? Wm1[r * 20 + c] : 0.f;
    b2s[idx] = bm2[idx];
  }
  if (t < 32) b1s[t] = (t < 20) ? bm1[t] : 0.f;
#if HAVE_TDM
  if (w == 0) {                    // 20x512 f32, fully contiguous 1D TDM copy
    tdm_load_2d(&W2[0][0], Wm2, 10240, 1, 10240, 0, 10240);
    TDM_WAIT();
  }
#else
  #pragma unroll
  for (int i = 0; i < 40; ++i) {
    int idx = t + 256 * i;
    ((float*)W2)[idx] = Wm2[idx];
  }
#endif
  __syncthreads();

  const int m0 = R0 + w * 16;
  v8f h0 = {}, h1 = {};
  #pragma unroll
  for (int c = 0; c < 4; ++c) {
    int kr = c * 4 + hi * 2;
    v2f a;   a.x   = cbuf[(size_t)(m0 + ln) * 16 + kr];
             a.y   = cbuf[(size_t)(m0 + ln) * 16 + kr + 1];
    v2f b0;  b0.x  = W1t[ln][kr];       b0.y  = W1t[ln][kr + 1];
    v2f b1v; b1v.x = W1t[ln + 16][kr];  b1v.y = W1t[ln + 16][kr + 1];
    h0 = wmma4(a, b0, h0);
    h1 = wmma4(a, b1v, h1);
  }
  float bb0 = b1s[ln], bb1 = b1s[ln + 16];
  #pragma unroll
  for (int r = 0; r < 8; ++r) {
    float f0 = h0[r] + bb0; f0 = f0 > 0.f ? f0 : 0.01f * f0;
    float f1 = h1[r] + bb1; f1 = f1 > 0.f ? f1 : 0.01f * f1;
    T[w][hi * 8 + r][ln]      = f0;
    T[w][hi * 8 + r][ln + 16] = f1;
  }
  asm volatile("s_wait_dscnt 0" ::: "memory");
  v2f ah[5];
  #pragma unroll
  for (int c = 0; c < 5; ++c) {    // K = 20 exactly
    int kr = c * 4 + hi * 2;
    ah[c].x = T[w][ln][kr]; ah[c].y = T[w][ln][kr + 1];
  }
  for (int nt = 0; nt < 32; ++nt) {
    v8f acc = {};
    #pragma unroll
    for (int c = 0; c < 5; ++c) {
      int kr = c * 4 + hi * 2;
      v2f bw; bw.x = W2[kr][nt * 16 + ln]; bw.y = W2[kr + 1][nt * 16 + ln];
      acc = wmma4(ah[c], bw, acc);
    }
    int col = nt * 16 + ln;
    float bb = b2s[col];
    #pragma unroll
    for (int r = 0; r < 8; ++r)
      out[(size_t)(m0 + hi * 8 + r) * 512 + col] = acc[r] + bb;
  }
}

extern "C" void kernel_launch(void* const* d_in, const int* in_sizes, int n_in,
                              void* d_out, int out_size, void* d_ws, size_t ws_size,
                              hipStream_t stream) {
  const float* x   = (const float*)d_in[0];
  const float* Wq  = (const float*)d_in[1];
  const float* bq  = (const float*)d_in[2];
  const float* Wk  = (const float*)d_in[3];
  const float* bk  = (const float*)d_in[4];
  const float* Wv  = (const float*)d_in[5];
  const float* bv  = (const float*)d_in[6];
  const float* Wm1 = (const float*)d_in[7];
  const float* bm1 = (const float*)d_in[8];
  const float* Wm2 = (const float*)d_in[9];
  const float* bm2 = (const float*)d_in[10];
  float* out = (float*)d_out;
  float* ws  = (float*)d_ws;

  float* qbuf = ws;                 // [32][2048][8]
  float* kbuf = ws + 524288;        // [32][2048][8]
  float* vbuf = ws + VOFF;          // [32][16][2048] dim-major (row5 = ones)
  float* cbuf = ws + 2097152;       // [16][2048][16]

  init_ws_kernel<<<(WS_FLOATS + 255) / 256, 256, 0, stream>>>(ws);
  qkv_kernel<<<256, 256, 0, stream>>>(x, Wq, bq, Wk, bk, Wv, bv, qbuf, kbuf, vbuf);
  attn_kernel<<<512, 256, 0, stream>>>(qbuf, kbuf, vbuf, cbuf);
  mlp_kernel<<<256, 256, 0, stream>>>(cbuf, Wm1, bm1, Wm2, bm2, out);
}